// AttentionLayer_9792525435162
// MI455X (gfx1250) — compile-verified
//
#include <hip/hip_runtime.h>
#include <math.h>

typedef __attribute__((ext_vector_type(2))) float v2f;
typedef __attribute__((ext_vector_type(8))) float v8f;
typedef __attribute__((ext_vector_type(4))) int   v4i;

#define DDIM 16   // capsule dim
#define KC   8    // capsules per node
#define MN   16   // neighbors per node
#define DTOT 128  // full feature dim

#define GAS __attribute__((address_space(1)))
#define LAS __attribute__((address_space(3)))

#if __has_builtin(__builtin_amdgcn_global_load_async_to_lds_b128)
#define HAVE_ASYNC_LDS 1
#else
#define HAVE_ASYNC_LDS 0
#endif

// jnp.maximum semantics: NaN in `a` propagates (fmaxf would drop it)
__device__ __forceinline__ float jmaxf(float a, float b) {
  return isnan(a) ? a : fmaxf(a, b);
}

__device__ __forceinline__ void wait_async_zero() {
#if HAVE_ASYNC_LDS
#if __has_builtin(__builtin_amdgcn_s_wait_asynccnt)
  __builtin_amdgcn_s_wait_asynccnt(0);
#else
  asm volatile("s_wait_asynccnt 0x0" ::: "memory");
#endif
#endif
}

__global__ __launch_bounds__(64)
void caps_route_kernel(const float* __restrict__ h,
                       const int*   __restrict__ neighbors,
                       const float* __restrict__ query,
                       const float* __restrict__ key_w,
                       const int*   __restrict__ iterat_p,
                       float* __restrict__ out_x,
                       float* __restrict__ out_att,
                       int N)
{
  // 2 waves per block, 2 nodes per wave -> 4 nodes per block
  __shared__ float sh_sub[4][17][DTOT];   // rows 0..15 = neighbors, row 16 = hc (34816 B)
  __shared__ float sh_nbsum[4][DTOT];     // 2048 B
  __shared__ float sh_red[2][256];        // per-wave 16x16 scratch (2048 B)
  __shared__ int   sh_idx[2][32];
  __shared__ float sh_inv[4][KC];
  __shared__ float sh_logit[4][KC];
  __shared__ float sh_att[4][KC];
  __shared__ float sh_diag[4][KC];
  __shared__ float sh_C[4][DDIM];

  const int tid  = threadIdx.x;
  const int wave = tid >> 5;
  const int lane = tid & 31;
  const int s0   = wave * 2;              // local node slot of this wave's first node
  const int g0   = blockIdx.x * 4 + s0;   // global node id of this wave's first node

  // ---- stage 1: load the two node rows (raw) into sub row 16; load neighbor ids ----
  {
    const float4* src0 = (const float4*)(h + (size_t)g0 * DTOT);
    const float4* src1 = (const float4*)(h + (size_t)(g0 + 1) * DTOT);
    ((float4*)sh_sub[s0][16])[lane]     = src0[lane];
    ((float4*)sh_sub[s0 + 1][16])[lane] = src1[lane];
  }
  int my_idx;
  {
    int j = lane >> 4;                    // node 0/1 within wave
    int m = lane & 15;                    // neighbor index
    my_idx = neighbors[(size_t)(g0 + j) * MN + m];
    sh_idx[wave][lane] = my_idx;
    __builtin_prefetch(h + (size_t)my_idx * DTOT, 0, 1);   // global_prefetch_b8
  }

#if HAVE_ASYNC_LDS
  // ---- stage 1b: issue the whole neighbor gather as async DMA (global -> LDS),
  //      overlapping with the hc / WMMA / softmax phase below. One b128 per lane
  //      per row: 32 lanes x 16B = one full 512B row per instruction, ASYNCcnt-tracked.
  #pragma unroll
  for (int r = 0; r < 32; ++r) {
    int idx_r = __shfl(my_idx, r, 32);    // row r's neighbor index (lane r holds it)
    int j = r >> 4, m = r & 15;
    uintptr_t gsrc = (uintptr_t)(h + (size_t)idx_r * DTOT + lane * 4);
    uintptr_t ldst = (uintptr_t)(&sh_sub[s0 + j][m][lane * 4]);
    __builtin_amdgcn_global_load_async_to_lds_b128(
        (GAS v4i*)gsrc, (LAS v4i*)ldst, 0, 0);
  }
#endif
  __syncthreads();

  // ---- stage 2: per-capsule L2 norms, then normalize row 16 in place (hc) ----
  if (lane < 16) {
    int s = s0 + (lane >> 3);
    int c = lane & 7;
    float ss = 0.f;
    #pragma unroll
    for (int i = 0; i < DDIM; ++i) { float v = sh_sub[s][16][c * DDIM + i]; ss += v * v; }
    sh_inv[s][c] = 1.0f / fmaxf(sqrtf(ss), 1e-12f);
  }
  __syncthreads();
  {
    int c = lane >> 2;                    // capsule of the 4 contiguous values this lane owns
    #pragma unroll
    for (int j = 0; j < 2; ++j) {
      float sc = sh_inv[s0 + j][c];
      #pragma unroll
      for (int u = 0; u < 4; ++u) sh_sub[s0 + j][16][lane * 4 + u] *= sc;
    }
  }
  __syncthreads();

  // ---- stage 3: WMMA  q = HC16x16 @ query,  kk = HC16x16 @ key_w ----
  // A (f32 16x4) layout: lanes 0-15 -> M=lane, K={4kb,4kb+1}; lanes 16-31 -> M=lane-16, K={4kb+2,4kb+3}
  // B (f32 4x16) layout mirrors A; C/D: lane = N (mod 16), VGPR v -> M = v + 8*(lane>=16)
  const int mRow   = lane & 15;
  const int hiHalf = lane >> 4;
  v8f cq, ck, cp;
  #pragma unroll
  for (int i = 0; i < 8; ++i) { cq[i] = 0.f; ck[i] = 0.f; }

  #pragma unroll
  for (int kb = 0; kb < 4; ++kb) {
    int ka = kb * 4 + hiHalf * 2;
    v2f a, bq, bk;
    // A element (m, f): rows 0-7 node s0, rows 8-15 node s0+1
    a.x = sh_sub[s0 + (mRow >> 3)][16][(mRow & 7) * DDIM + ka];
    a.y = sh_sub[s0 + (mRow >> 3)][16][(mRow & 7) * DDIM + ka + 1];
    bq.x = query[ka * DDIM + mRow];        bq.y = query[(ka + 1) * DDIM + mRow];
    bk.x = key_w[ka * DDIM + mRow];        bk.y = key_w[(ka + 1) * DDIM + mRow];
    cq = __builtin_amdgcn_wmma_f32_16x16x4_f32(false, a, false, bq, (short)0, cq, false, false);
    ck = __builtin_amdgcn_wmma_f32_16x16x4_f32(false, a, false, bk, (short)0, ck, false, false);
  }

  // ---- stage 4: att_logit[k] = sum_f q[k,f]*kk[k,f]  (row-reduce of elementwise product) ----
  #pragma unroll
  for (int v = 0; v < 8; ++v) {
    int Mr = v + hiHalf * 8;
    sh_red[wave][Mr * 16 + mRow] = cq[v] * ck[v];
  }
  __syncthreads();
  if (lane < 16) {
    float ssum = 0.f;
    #pragma unroll
    for (int nn = 0; nn < 16; ++nn) ssum += sh_red[wave][lane * 16 + nn];
    sh_logit[s0 + (lane >> 3)][lane & 7] = ssum;
  }
  __syncthreads();
  if (lane < 2) {                          // softmax over 8 capsules, one node per lane
    int s = s0 + lane, g = g0 + lane;
    float mx = -INFINITY;
    #pragma unroll
    for (int k = 0; k < KC; ++k) mx = fmaxf(mx, sh_logit[s][k]);
    float e[KC]; float ssum = 0.f;
    #pragma unroll
    for (int k = 0; k < KC; ++k) { e[k] = expf(sh_logit[s][k] - mx); ssum += e[k]; }
    #pragma unroll
    for (int k = 0; k < KC; ++k) {
      float a = e[k] / ssum;
      sh_att[s][k] = a;
      out_att[(size_t)g * KC + k] = a;
    }
  }

  // ---- stage 5: make the gathered neighbor rows visible ----
#if HAVE_ASYNC_LDS
  wait_async_zero();                       // all 32 row-DMAs for this wave done
#else
  for (int r = 0; r < 32; ++r) {           // fallback: VGPR round-trip gather
    int j = r >> 4, m = r & 15;
    int idx = sh_idx[wave][r];
    const float4* src = (const float4*)(h + (size_t)idx * DTOT);
    ((float4*)sh_sub[s0 + j][m])[lane] = src[lane];
  }
#endif
  __syncthreads();

  // ---- stage 6: nbsum over neighbors (8 columns per lane) ----
  #pragma unroll
  for (int u = 0; u < 8; ++u) {
    int c = lane * 8 + u;                  // 0..255
    int s = s0 + (c >> 7);
    int col = c & 127;
    float acc = 0.f;
    #pragma unroll
    for (int r = 0; r < MN; ++r) acc += sh_sub[s][r][col];
    sh_nbsum[s][col] = acc;
  }
  __syncthreads();

  // ---- stage 7: WMMA  qs_sum = NBSUM16x16 @ query; diagv[k] = sum_f qs_sum[k,f]*kk[k,f] ----
  #pragma unroll
  for (int i = 0; i < 8; ++i) cp[i] = 0.f;
  #pragma unroll
  for (int kb = 0; kb < 4; ++kb) {
    int ka = kb * 4 + hiHalf * 2;
    v2f a, bq;
    a.x = sh_nbsum[s0 + (mRow >> 3)][(mRow & 7) * DDIM + ka];
    a.y = sh_nbsum[s0 + (mRow >> 3)][(mRow & 7) * DDIM + ka + 1];
    bq.x = query[ka * DDIM + mRow];        bq.y = query[(ka + 1) * DDIM + mRow];
    cp = __builtin_amdgcn_wmma_f32_16x16x4_f32(false, a, false, bq, (short)0, cp, false, false);
  }
  #pragma unroll
  for (int v = 0; v < 8; ++v) {
    int Mr = v + hiHalf * 8;
    sh_red[wave][Mr * 16 + mRow] = cp[v] * ck[v];
  }
  __syncthreads();
  if (lane < 16) {
    float ssum = 0.f;
    #pragma unroll
    for (int nn = 0; nn < 16; ++nn) ssum += sh_red[wave][lane * 16 + nn];
    sh_diag[s0 + (lane >> 3)][lane & 7] = ssum;
  }
  __syncthreads();

  // ---- stage 8: per-column lower-median of 17 normalized values; C[d] contributions ----
  #pragma unroll 1
  for (int u = 0; u < 8; ++u) {
    int c = lane * 8 + u;
    int s = s0 + (c >> 7);
    int col = c & 127;
    int cj = col >> 4;                     // capsule index j
    float v[17];
    float ss = 0.f;
    #pragma unroll
    for (int i = 0; i < 17; ++i) { v[i] = sh_sub[s][i][col]; ss += v[i] * v[i]; }
    float inv = 1.0f / fmaxf(sqrtf(ss), 1e-12f);
    // lower median = stable-sorted element at index 8 of 17
    float med = 0.f;
    #pragma unroll
    for (int i = 0; i < 17; ++i) {
      int r = 0;
      #pragma unroll
      for (int j2 = 0; j2 < 17; ++j2) {
        r += (v[j2] < v[i]) ? 1 : 0;
        r += ((j2 < i) && (v[j2] == v[i])) ? 1 : 0;
      }
      if (r == 8) med = v[i];
    }
    float middle = med * inv;              // normalization is order-preserving
    float d0 = v[16] - middle;             // hc - middle  (row 16 already normalized)
    sh_red[wave][c] = sh_diag[s][cj] * d0 * d0;
  }
  __syncthreads();
  {
    int s = s0 + (lane >> 4);
    int d = lane & 15;
    float acc = 0.f;
    #pragma unroll
    for (int j2 = 0; j2 < 8; ++j2) acc += sh_red[wave][(lane >> 4) * 128 + j2 * 16 + d];
    sh_C[s][d] = acc;                      // cov rows are identical -> single C[d] per node
  }
  __syncthreads();

  // ---- stage 9: serial routing (tiny), one node per lane 0/16 ----
  if ((lane & 15) == 0) {
    int nl = lane >> 4;
    int s = s0 + nl;
    int g = g0 + nl;
    // left is identical across capsules (cov rows identical)
    float lsum = 0.f;
    #pragma unroll
    for (int d = 0; d < DDIM; ++d) {
      float lr = logf(sh_C[s][d]);
      if (isnan(lr)) lr = 1e-6f;
      lsum += logf(lr);                    // double log, NaN/Inf flow as in reference
    }
    float dets = expf(lsum);
    const float coef = 5.822045e-07f;      // 1/sqrt((2*pi)^16 / 2)
    float left = coef / sqrtf(dets + 1e-6f);

    float ix[DDIM];
    #pragma unroll
    for (int d = 0; d < DDIM; ++d) {
      float a = 0.f;
      #pragma unroll
      for (int k = 0; k < KC; ++k) a += sh_att[s][k] * sh_sub[s][16][k * DDIM + d];
      ix[d] = a;
    }
    int iters = *iterat_p;
    for (int it = 0; it < iters; ++it) {
      float p[KC]; float pn = 0.f;
      #pragma unroll
      for (int k = 0; k < KC; ++k) {
        float rs = 0.f;
        #pragma unroll
        for (int d = 0; d < DDIM; ++d) {
          float df = ix[d] - sh_sub[s][16][k * DDIM + d];
          rs += sh_C[s][d] * df * df;
        }
        p[k] = left * expf(-0.5f * rs);
        pn += p[k] * p[k];
      }
      float den = jmaxf(sqrtf(pn), 1e-12f);
      float wp[KC]; float s1 = 0.f;
      #pragma unroll
      for (int k = 0; k < KC; ++k) {
        float pk = p[k] / den;
        if (isnan(pk)) pk = 1e-6f;
        wp[k] = sh_att[s][k] * pk;
        s1 += wp[k];
      }
      #pragma unroll
      for (int d = 0; d < DDIM; ++d) {
        float a = 0.f;
        #pragma unroll
        for (int k = 0; k < KC; ++k) a += wp[k] * sh_sub[s][16][k * DDIM + d];
        ix[d] = a / (s1 + 1e-9f);
      }
    }
    float nn2 = 0.f;
    #pragma unroll
    for (int d = 0; d < DDIM; ++d) nn2 += ix[d] * ix[d];
    float den = jmaxf(sqrtf(nn2), 1e-12f);
    #pragma unroll
    for (int d = 0; d < DDIM; ++d) out_x[(size_t)g * DDIM + d] = ix[d] / den;
  }
}

extern "C" void kernel_launch(void* const* d_in, const int* in_sizes, int n_in,
                              void* d_out, int out_size, void* d_ws, size_t ws_size,
                              hipStream_t stream) {
  const float* h         = (const float*)d_in[0];
  const int*   neighbors = (const int*)d_in[1];
  const float* query     = (const float*)d_in[2];
  const float* key_w     = (const float*)d_in[3];
  const int*   iterat    = (const int*)d_in[4];
  (void)n_in; (void)d_ws; (void)ws_size; (void)out_size;

  int N = in_sizes[0] / DTOT;              // 65536
  float* out_x   = (float*)d_out;          // [N,16]
  float* out_att = out_x + (size_t)N * DDIM; // [N,8]

  int blocks = N / 4;                      // 4 nodes per 64-thread block
  hipLaunchKernelGGL(caps_route_kernel, dim3(blocks), dim3(64), 0, stream,
                     h, neighbors, query, key_w, iterat, out_x, out_att, N);
}